// SparseResidualBlock_33638183862316
// MI455X (gfx1250) — compile-verified
//
#include <hip/hip_runtime.h>

typedef __attribute__((ext_vector_type(2))) float v2f;
typedef __attribute__((ext_vector_type(4))) float v4f;
typedef __attribute__((ext_vector_type(8))) float v8f;

#define CCH 32          // channels
#define KT 9            // kernel taps
#define WAVES 8         // waves per block (wave32)
#define ROWS_PER_BLOCK 128
#define GSTRIDE 36      // LDS row stride in floats (16B-aligned rows, conflict-free)

// One wave computes a 16-row x 32-col output tile:
//   out[16x32] = sum_k (masked gather of src rows)[16x32] @ W[k][32x32]
// via chained V_WMMA_F32_16X16X4_F32 (8 K-steps per tap, 2 N-tiles).
// Gathered rows are copied memory->LDS with GLOBAL_LOAD_ASYNC_TO_LDS_B128
// (ASYNCcnt-tracked, double buffered per wave -> no barriers in the tap loop).
__global__ __launch_bounds__(256) void subm_conv_wmma(
    const float* __restrict__ src, const float* __restrict__ W,
    const int* __restrict__ nbr_idx, const int* __restrict__ nbr_mask,
    const float* __restrict__ bnscale, const float* __restrict__ bnshift, int fuse,
    float* __restrict__ out, float* __restrict__ psum, float* __restrict__ psq,
    int n)
{
    // Weights pre-swizzled into B-fragment order: one float2 per (k,s,t,lane).
    __shared__ __align__(16) float wlds[KT * 8 * 2 * 32 * 2];        // 36 KB
    __shared__ __align__(16) float gath[2 * WAVES * 16 * GSTRIDE];   // 36 KB, dbl-buf
    __shared__ float lsum[WAVES * 2 * 32];
    __shared__ float lsq [WAVES * 2 * 32];

    const int tid  = threadIdx.x;
    const int wave = tid >> 5;
    const int lane = tid & 31;
    const int half = lane >> 4;   // 0: K/M low half, 1: high half
    const int l16  = lane & 15;

    // ---- Stage W into LDS in WMMA-B fragment layout ----
    // B frag for (k, s, t): lane needs W[k][4s+2*half + {0,1}][16t + l16] as float2.
    v2f* wp = (v2f*)wlds;
    for (int f = tid; f < KT * 8 * 2 * 32; f += 256) {
        int ln = f & 31, t = (f >> 5) & 1, s = (f >> 6) & 7, k = f >> 9;
        int c0  = 4 * s + 2 * (ln >> 4);
        int col = 16 * t + (ln & 15);
        const float* wb = W + k * (CCH * CCH) + c0 * CCH + col;
        v2f w; w.x = wb[0]; w.y = wb[CCH];
        wp[f] = w;
    }

    // Per-lane BN params for the A-fragment channel pair of each K-step:
    // step s touches channels (4s + 2*half, 4s + 2*half + 1).
    v2f scp[8], shp[8];
    if (fuse) {
        #pragma unroll
        for (int s = 0; s < 8; ++s) {
            scp[s] = *(const v2f*)(bnscale + 4 * s + 2 * half);
            shp[s] = *(const v2f*)(bnshift + 4 * s + 2 * half);
        }
    }

    const int  row_base = blockIdx.x * ROWS_PER_BLOCK + wave * 16;
    const int  row_g    = row_base + l16;     // row this lane stages
    const bool valid    = row_g < n;

    float* gbuf0 = gath + wave * 16 * GSTRIDE;
    float* gbuf1 = gath + (WAVES + wave) * 16 * GSTRIDE;
    // LDS byte addresses (wave-relative: low 32 bits of the flat LDS pointer)
    unsigned ldsa0 = (unsigned)(uintptr_t)(gbuf0 + l16 * GSTRIDE + half * 16);
    unsigned ldsa1 = (unsigned)(uintptr_t)(gbuf1 + l16 * GSTRIDE + half * 16);

    // Issue the 4 async 16B copies that stage this lane's half-row of tap k.
    // INST_OFFSET advances BOTH the LDS and the global address (ISA 10.x).
    auto issue = [&](int k, unsigned ldsa, float& mout) {
        int idx = 0, mi = 0;
        if (valid) {
            idx = nbr_idx [(size_t)k * n + row_g];
            mi  = nbr_mask[(size_t)k * n + row_g];
        }
        mout = valid ? (float)mi : 0.0f;
        unsigned voff = (unsigned)idx * (CCH * 4) + (unsigned)(half * 64);
        asm volatile(
            "s_wait_dscnt 0x0\n\t"
            "global_load_async_to_lds_b128 %0, %1, %2\n\t"
            "global_load_async_to_lds_b128 %0, %1, %2 offset:16\n\t"
            "global_load_async_to_lds_b128 %0, %1, %2 offset:32\n\t"
            "global_load_async_to_lds_b128 %0, %1, %2 offset:48"
            :: "v"(ldsa), "v"(voff), "s"(src) : "memory");
    };

    __syncthreads();   // wlds visible to all waves

    float m_cur, m_next;
    issue(0, ldsa0, m_cur);

    v8f acc0 = {}; v8f acc1 = {};

    for (int k = 0; k < KT; ++k) {
        const float* g = (k & 1) ? gbuf1 : gbuf0;

        if (k + 1 < KT) {
            issue(k + 1, (k & 1) ? ldsa0 : ldsa1, m_next);
            // 4 outstanding for tap k+1; tap k's 4 complete in order first.
            asm volatile("s_wait_asynccnt 0x4" ::: "memory");
        } else {
            m_next = 0.0f;
            asm volatile("s_wait_asynccnt 0x0" ::: "memory");
        }

        // ---- 8 K-steps of fp32 WMMA, 2 output column tiles ----
        #pragma unroll
        for (int s = 0; s < 8; ++s) {
            // A 16x4 f32 frag: lane<16 -> K = 4s,4s+1 ; lane>=16 -> K = 4s+2,4s+3
            v2f a = *(const v2f*)(g + l16 * GSTRIDE + 4 * s + 2 * half);
            if (fuse) {   // BN1 + ReLU fused on the gathered row
                a = a * scp[s] + shp[s];
                a.x = fmaxf(a.x, 0.0f);
                a.y = fmaxf(a.y, 0.0f);
            }
            a *= m_cur;   // rulebook mask (per gathered row)
            v2f b0 = wp[((k * 8 + s) * 2 + 0) * 32 + lane];
            v2f b1 = wp[((k * 8 + s) * 2 + 1) * 32 + lane];
            acc0 = __builtin_amdgcn_wmma_f32_16x16x4_f32(
                false, a, false, b0, (short)0, acc0, false, false);
            acc1 = __builtin_amdgcn_wmma_f32_16x16x4_f32(
                false, a, false, b1, (short)0, acc1, false, false);
        }
        m_cur = m_next;
    }

    // ---- Store outputs + per-lane BN partial stats ----
    // D layout: VGPR r, lanes 0-15 -> M=r, lanes 16-31 -> M=8+r; N = l16 (+16 for acc1)
    float s0 = 0.f, q0 = 0.f, s1 = 0.f, q1 = 0.f;
    #pragma unroll
    for (int r = 0; r < 8; ++r) {
        int row = row_base + 8 * half + r;
        float v0 = acc0[r], v1 = acc1[r];
        s0 += v0; q0 += v0 * v0;
        s1 += v1; q1 += v1 * v1;
        if (row < n) {
            out[(size_t)row * CCH + l16]      = v0;
            out[(size_t)row * CCH + 16 + l16] = v1;
        }
    }
    lsum[(wave * 2 + half) * 32 + l16]      = s0;
    lsum[(wave * 2 + half) * 32 + 16 + l16] = s1;
    lsq [(wave * 2 + half) * 32 + l16]      = q0;
    lsq [(wave * 2 + half) * 32 + 16 + l16] = q1;
    __syncthreads();

    // Fixed-order block reduction (deterministic; no float atomics)
    if (tid < 32) {
        float S = 0.f, Q = 0.f;
        for (int w = 0; w < WAVES * 2; ++w) {
            S += lsum[w * 32 + tid];
            Q += lsq [w * 32 + tid];
        }
        psum[(size_t)blockIdx.x * 32 + tid] = S;
        psq [(size_t)blockIdx.x * 32 + tid] = Q;
    }
}

__global__ void bn_finalize(const float* __restrict__ psum, const float* __restrict__ psq,
                            const float* __restrict__ gamma, const float* __restrict__ beta,
                            float* __restrict__ scale, float* __restrict__ shift,
                            int nblocks, float invN)
{
    int c = threadIdx.x;
    if (c >= 32) return;
    float S = 0.f, Q = 0.f;
    for (int b = 0; b < nblocks; ++b) {   // fixed order -> deterministic
        S += psum[(size_t)b * 32 + c];
        Q += psq [(size_t)b * 32 + c];
    }
    float mean = S * invN;
    float var  = Q * invN - mean * mean;
    float rstd = rsqrtf(var + 1e-4f);
    float sccv = rstd * gamma[c];
    scale[c] = sccv;
    shift[c] = beta[c] - mean * sccv;
}

__global__ void epilogue(const float* __restrict__ h2, const float* __restrict__ x,
                         const float* __restrict__ scale, const float* __restrict__ shift,
                         float* __restrict__ out, int total)
{
    int i = blockIdx.x * blockDim.x + threadIdx.x;
    if (i >= total) return;
    int c = i & 31;
    float v = h2[i] * scale[c] + shift[c] + x[i];
    out[i] = fmaxf(v, 0.f);
}

extern "C" void kernel_launch(void* const* d_in, const int* in_sizes, int n_in,
                              void* d_out, int out_size, void* d_ws, size_t ws_size,
                              hipStream_t stream) {
    const float* x      = (const float*)d_in[0];
    const float* W1     = (const float*)d_in[1];
    const float* gamma1 = (const float*)d_in[2];
    const float* beta1  = (const float*)d_in[3];
    const float* W2     = (const float*)d_in[4];
    const float* gamma2 = (const float*)d_in[5];
    const float* beta2  = (const float*)d_in[6];
    const int* nbr_idx  = (const int*)d_in[7];
    const int* nbr_mask = (const int*)d_in[8];
    float* out = (float*)d_out;

    const int n = in_sizes[0] / CCH;
    const int nblocks = (n + ROWS_PER_BLOCK - 1) / ROWS_PER_BLOCK;

    // Workspace layout
    float* ws     = (float*)d_ws;
    float* h2     = ws;                                    // n*32 floats (64 MB)
    float* psum   = ws + (size_t)n * CCH;
    float* psq    = psum + (size_t)nblocks * CCH;
    float* scale1 = psq  + (size_t)nblocks * CCH;
    float* shift1 = scale1 + CCH;
    float* scale2 = shift1 + CCH;
    float* shift2 = scale2 + CCH;

    dim3 grid(nblocks), block(256);
    const float invN = 1.0f / (float)n;

    // conv1: h1 -> d_out, BN1 partials
    subm_conv_wmma<<<grid, block, 0, stream>>>(
        x, W1, nbr_idx, nbr_mask, nullptr, nullptr, 0, out, psum, psq, n);
    bn_finalize<<<1, 32, 0, stream>>>(psum, psq, gamma1, beta1, scale1, shift1, nblocks, invN);

    // conv2 with BN1+ReLU fused into the A-fragment path: h2 -> ws, BN2 partials
    subm_conv_wmma<<<grid, block, 0, stream>>>(
        out, W2, nbr_idx, nbr_mask, scale1, shift1, 1, h2, psum, psq, n);
    bn_finalize<<<1, 32, 0, stream>>>(psum, psq, gamma2, beta2, scale2, shift2, nblocks, invN);

    // out = relu(bn2(h2) + x)
    const int total = n * CCH;
    epilogue<<<(total + 255) / 256, 256, 0, stream>>>(h2, x, scale2, shift2, out, total);
}